// QuantumLayer_12060268167387
// MI455X (gfx1250) — compile-verified
//
#include <hip/hip_runtime.h>
#include <hip/hip_bf16.h>

typedef __attribute__((ext_vector_type(2))) float v2f;
typedef __attribute__((ext_vector_type(4))) float v4f;
typedef __attribute__((ext_vector_type(8))) float v8f;
typedef __attribute__((ext_vector_type(4))) unsigned int u32x4;
typedef __attribute__((ext_vector_type(4))) int i32x4;
typedef __attribute__((ext_vector_type(8))) int i32x8;

#define NQ    6
#define DIMQ  64
#define D     512

#if defined(__has_builtin)
# if __has_builtin(__builtin_amdgcn_tensor_load_to_lds) && __has_builtin(__builtin_amdgcn_s_wait_tensorcnt)
#  define HAVE_TDM 1
# endif
#endif

// ------------------------------------------------------------------
// Setup: build the fixed 64x64 entangler unitary U from `weights`,
// fold the (-i)^popcount(col) embedding phase in, store Re/Im row-major.
// ------------------------------------------------------------------
__global__ __launch_bounds__(64) void qc_setup_kernel(const float* __restrict__ weights,
                                                      float* __restrict__ Rws,
                                                      float* __restrict__ Iws) {
  __shared__ float2 st[64][65];
  const int t = threadIdx.x;                 // column index
  for (int r = 0; r < DIMQ; ++r) st[t][r] = make_float2(r == t ? 1.f : 0.f, 0.f);

  for (int l = 0; l < 2; ++l) {
    for (int q = 0; q < NQ; ++q) {
      const float phi = weights[(l*NQ + q)*3 + 0];
      const float th  = weights[(l*NQ + q)*3 + 1];
      const float om  = weights[(l*NQ + q)*3 + 2];
      float sn, ct; sincosf(0.5f*th, &sn, &ct);
      float sa, ca; sincosf(0.5f*(phi+om), &sa, &ca);   // ep = ca - i*sa
      float sb, cb; sincosf(0.5f*(phi-om), &sb, &cb);   // em = cb + i*sb
      const float2 u00 = make_float2( ca*ct, -sa*ct);
      const float2 u01 = make_float2(-cb*sn, -sb*sn);
      const float2 u10 = make_float2( cb*sn, -sb*sn);
      const float2 u11 = make_float2( ca*ct,  sa*ct);
      const int mask = 1 << (5 - q);                    // qubit 0 = MSB
      for (int i = 0; i < DIMQ; ++i) {
        if (i & mask) continue;
        const int j = i | mask;
        const float2 a = st[t][i], b = st[t][j];
        st[t][i] = make_float2(u00.x*a.x - u00.y*a.y + u01.x*b.x - u01.y*b.y,
                               u00.x*a.y + u00.y*a.x + u01.x*b.y + u01.y*b.x);
        st[t][j] = make_float2(u10.x*a.x - u10.y*a.y + u11.x*b.x - u11.y*b.y,
                               u10.x*a.y + u10.y*a.x + u11.x*b.y + u11.y*b.x);
      }
    }
    const int rr = l + 1;
    for (int q = 0; q < NQ; ++q) {
      const int mc = 1 << (5 - q), mt = 1 << (5 - ((q + rr) % NQ));
      for (int i = 0; i < DIMQ; ++i) {
        if ((i & mc) && !(i & mt)) {
          const int j = i | mt;
          const float2 tmp = st[t][i]; st[t][i] = st[t][j]; st[t][j] = tmp;
        }
      }
    }
  }
  const int p = __popc(t) & 3;
  for (int r = 0; r < DIMQ; ++r) {
    const float2 v = st[t][r];
    float re, im;
    if      (p == 0) { re =  v.x; im =  v.y; }
    else if (p == 1) { re =  v.y; im = -v.x; }
    else if (p == 2) { re = -v.x; im = -v.y; }
    else             { re = -v.y; im =  v.x; }
    Rws[r*DIMQ + t] = re;
    Iws[r*DIMQ + t] = im;
  }
}

// ------------------------------------------------------------------
// Main fused kernel: 8 waves/block, 16 batch rows per wave.
// ------------------------------------------------------------------
#define BLK_THREADS    256
#define WAVES          8
#define ROWS_PER_BLOCK 128

#define W_STRIDE  516
#define A_STRIDE  68
#define X_STRIDE  132
#define OFF_W     0
#define OFF_R     (OFF_W + 16*W_STRIDE)
#define OFF_I     (OFF_R + DIMQ*A_STRIDE)
#define OFF_X     (OFF_I + DIMQ*A_STRIDE)
#define X_PER_SL  (16*X_STRIDE)                 // 2112 floats per slice
#define X_PER_W   (2*X_PER_SL)                  // double-buffered for TDM
#define OFF_Z     (OFF_X + WAVES*X_PER_W)
#define Z_PER_W   256
#define OFF_M     (OFF_Z + WAVES*Z_PER_W)
#define M_PER_W   (16*A_STRIDE)
#define OFF_P     (OFF_M + WAVES*M_PER_W)
#define LDS_FLOATS (OFF_P + WAVES*M_PER_W)      // 70208 floats = 280,832 B (< 320 KB)

#define WMMA_F32(A, B, C) \
  __builtin_amdgcn_wmma_f32_16x16x4_f32(false, (A), false, (B), (short)0, (C), false, false)

#if defined(HAVE_TDM)
// One TDM descriptor moves a 16x128-f32 tile (row stride 512 floats in memory)
// into LDS with a 4-dword pad after each 128-dword row (-> 132-float stride).
__device__ __forceinline__ void tdm_tile_load(unsigned long long ga, unsigned lds_byte) {
  const u32x4 g0 = { 1u,                                   // count=1, user mode
                     lds_byte,                             // lds_addr
                     (unsigned)ga,                         // global_addr[31:0]
                     (unsigned)((ga >> 32) & 0x01FFFFFFu) | (2u << 30) }; // ga[56:32] | type=2
  const i32x8 g1 = { (int)((2u<<16) | (1u<<20) | (6u<<22) | (3u<<25)),
                     // data_size=4B, pad_enable, pad_interval=128dw, pad_amount=4dw
                     (int)(512u << 16),        // tensor_dim0 = 512 (lo16 in bits 63:48)
                     0,                        // tensor_dim0 hi16 = 0, tensor_dim1 lo16 = 0
                     (int)(1u | (128u << 16)), // tensor_dim1 hi16 = 1 (=65536), tile_dim0 = 128
                     16,                       // tile_dim1 = 16, tile_dim2 = 0
                     512,                      // tensor_dim0_stride = 512
                     0, 0 };
  const i32x4 z4 = { 0, 0, 0, 0 };
#if __clang_major__ >= 23
  const i32x8 z8 = { 0, 0, 0, 0, 0, 0, 0, 0 };
  __builtin_amdgcn_tensor_load_to_lds(g0, g1, z4, z4, z8, 0);
#else
  __builtin_amdgcn_tensor_load_to_lds(g0, g1, z4, z4, 0);
#endif
}
#endif

__global__ __launch_bounds__(BLK_THREADS) void qc_main_kernel(
    const float* __restrict__ x, const float* __restrict__ Wm,
    const float* __restrict__ bias,
    const float* __restrict__ Rws, const float* __restrict__ Iws,
    float* __restrict__ out) {
  extern __shared__ float lds[];
  const int tid   = threadIdx.x;
  const int wave  = tid >> 5;
  const int lane  = tid & 31;
  const int n16   = lane & 15;      // A: row M ; B: col N
  const int khalf = lane >> 4;      // K-slot half within a WMMA

  // ---- cooperative staging: W (zero-padded 6->16 rows), R, I ----
  for (int idx = tid; idx < 16*D; idx += BLK_THREADS) {
    const int r = idx >> 9, c = idx & (D - 1);
    lds[OFF_W + r*W_STRIDE + c] = (r < NQ) ? Wm[r*D + c] : 0.f;
  }
  for (int idx = tid; idx < DIMQ*DIMQ; idx += BLK_THREADS) {
    const int r = idx >> 6, c = idx & 63;
    lds[OFF_R + r*A_STRIDE + c] = Rws[idx];
    lds[OFF_I + r*A_STRIDE + c] = Iws[idx];
  }
  __syncthreads();

  float* xw = lds + OFF_X + wave*X_PER_W;     // two slices of X_PER_SL
  float* zw = lds + OFF_Z + wave*Z_PER_W;
  float* mw = lds + OFF_M + wave*M_PER_W;
  float* pw = lds + OFF_P + wave*M_PER_W;

  const long rowbase = (long)blockIdx.x*ROWS_PER_BLOCK + wave*16;

  // ---- stage 1: z(16x16) = x(16x512) @ W^T via V_WMMA_F32_16X16X4_F32 ----
  // K processed in a permuted order so each lane's A/B data for two consecutive
  // WMMAs is one contiguous float4 (single b128 LDS read each).
  v8f acc0 = {}, acc1 = {};
#if defined(HAVE_TDM)
  {
    const unsigned long long gbase = (unsigned long long)(uintptr_t)(x + rowbase*D);
    const unsigned l0 = (unsigned)(uintptr_t)(void*)xw;            // LDS byte addr = low 32 bits
    const unsigned l1 = (unsigned)(uintptr_t)(void*)(xw + X_PER_SL);
    tdm_tile_load(gbase, l0);                                      // chunk 0
    #pragma unroll
    for (int kc = 0; kc < 4; ++kc) {
      if (kc < 3) {
        tdm_tile_load(gbase + (unsigned)(kc + 1)*512u,             // +128 floats
                      (kc & 1) ? l0 : l1);
        __builtin_amdgcn_s_wait_tensorcnt(1);                      // chunk kc has landed
      } else {
        __builtin_amdgcn_s_wait_tensorcnt(0);
      }
      const float* xs = (kc & 1) ? (xw + X_PER_SL) : xw;
      const int kbase = kc*128;
      #pragma unroll
      for (int jp = 0; jp < 16; ++jp) {
        const int cc = 8*jp + 4*khalf;
        const v4f a4 = *(const v4f*)(xs + n16*X_STRIDE + cc);
        const v4f b4 = *(const v4f*)(lds + OFF_W + n16*W_STRIDE + kbase + cc);
        acc0 = WMMA_F32(__builtin_shufflevector(a4, a4, 0, 1),
                        __builtin_shufflevector(b4, b4, 0, 1), acc0);
        acc1 = WMMA_F32(__builtin_shufflevector(a4, a4, 2, 3),
                        __builtin_shufflevector(b4, b4, 2, 3), acc1);
      }
    }
  }
#else
  {
    v4f buf[2][16];
    #pragma unroll
    for (int it = 0; it < 16; ++it)
      buf[0][it] = __builtin_nontemporal_load((const v4f*)(x + (rowbase + it)*D + lane*4));
    #pragma unroll
    for (int kc = 0; kc < 4; ++kc) {
      const int kbase = kc*128;
      #pragma unroll
      for (int it = 0; it < 16; ++it)
        *(v4f*)(xw + it*X_STRIDE + lane*4) = buf[kc & 1][it];
      if (kc < 3) {
        #pragma unroll
        for (int it = 0; it < 16; ++it)
          buf[(kc + 1) & 1][it] = __builtin_nontemporal_load(
              (const v4f*)(x + (rowbase + it)*D + kbase + 128 + lane*4));
      }
      #pragma unroll
      for (int jp = 0; jp < 16; ++jp) {
        const int cc = 8*jp + 4*khalf;
        const v4f a4 = *(const v4f*)(xw + n16*X_STRIDE + cc);
        const v4f b4 = *(const v4f*)(lds + OFF_W + n16*W_STRIDE + kbase + cc);
        acc0 = WMMA_F32(__builtin_shufflevector(a4, a4, 0, 1),
                        __builtin_shufflevector(b4, b4, 0, 1), acc0);
        acc1 = WMMA_F32(__builtin_shufflevector(a4, a4, 2, 3),
                        __builtin_shufflevector(b4, b4, 2, 3), acc1);
      }
    }
  }
#endif
  const v8f acc = acc0 + acc1;
  #pragma unroll
  for (int v = 0; v < 8; ++v) zw[(v + 8*khalf)*16 + n16] = acc[v];   // z[m][n]

  // ---- stage 2: bias + normalize + product-state magnitudes M[16][64] ----
  {
    float zt[NQ], cq[NQ], sq[NQ], ss = 0.f;
    #pragma unroll
    for (int q = 0; q < NQ; ++q) { zt[q] = zw[n16*16 + q] + bias[q]; ss += zt[q]*zt[q]; }
    const float inv = rsqrtf(ss);
    #pragma unroll
    for (int q = 0; q < NQ; ++q) sincosf(0.5f*zt[q]*inv, &sq[q], &cq[q]);
    const int ibase = khalf*32;
    for (int ii = 0; ii < 32; ++ii) {
      const int i = ibase + ii;
      float m = 1.f;
      #pragma unroll
      for (int q = 0; q < NQ; ++q) m *= ((i >> (5 - q)) & 1) ? sq[q] : cq[q];
      mw[n16*A_STRIDE + i] = m;
    }
  }

  // ---- stage 3: P = (M R^T)^2 + (M I^T)^2 via WMMA (2 x 16x64x64) ----
  #pragma unroll
  for (int sub = 0; sub < 4; ++sub) {
    v8f yr = {}, yi = {};
    #pragma unroll
    for (int jp = 0; jp < 8; ++jp) {
      const int cc = 8*jp + 4*khalf;
      const v4f a4 = *(const v4f*)(mw + n16*A_STRIDE + cc);
      const v4f r4 = *(const v4f*)(lds + OFF_R + (sub*16 + n16)*A_STRIDE + cc);
      const v4f i4 = *(const v4f*)(lds + OFF_I + (sub*16 + n16)*A_STRIDE + cc);
      const v2f a01 = __builtin_shufflevector(a4, a4, 0, 1);
      const v2f a23 = __builtin_shufflevector(a4, a4, 2, 3);
      yr = WMMA_F32(a01, __builtin_shufflevector(r4, r4, 0, 1), yr);
      yi = WMMA_F32(a01, __builtin_shufflevector(i4, i4, 0, 1), yi);
      yr = WMMA_F32(a23, __builtin_shufflevector(r4, r4, 2, 3), yr);
      yi = WMMA_F32(a23, __builtin_shufflevector(i4, i4, 2, 3), yi);
    }
    #pragma unroll
    for (int v = 0; v < 8; ++v)
      pw[(v + 8*khalf)*A_STRIDE + sub*16 + n16] = yr[v]*yr[v] + yi[v]*yi[v];
  }

  // ---- stage 4: out[m][q] = sum_r (1 - 2*bit_q(r)) * P[m][r] ----
  for (int o = lane; o < 16*NQ; o += 32) {
    const int m = o / NQ, q = o % NQ;
    float a0 = 0.f;
    for (int r = 0; r < DIMQ; ++r) {
      const float sgn = ((r >> (5 - q)) & 1) ? -1.f : 1.f;
      a0 = fmaf(sgn, pw[m*A_STRIDE + r], a0);
    }
    out[(rowbase + m)*NQ + q] = a0;
  }
}

// ------------------------------------------------------------------
extern "C" void kernel_launch(void* const* d_in, const int* in_sizes, int n_in,
                              void* d_out, int out_size, void* d_ws, size_t ws_size,
                              hipStream_t stream) {
  const float* x       = (const float*)d_in[0];   // (B, 512)
  const float* Wm      = (const float*)d_in[1];   // (6, 512)
  const float* b       = (const float*)d_in[2];   // (6,)
  const float* weights = (const float*)d_in[3];   // (2, 6, 3)
  float* out = (float*)d_out;                     // (B, 6)

  float* Rws = (float*)d_ws;                      // 64*64 f32
  float* Iws = Rws + DIMQ*DIMQ;                   // 64*64 f32

  const int Brows = in_sizes[0] / D;              // 65536
  qc_setup_kernel<<<1, 64, 0, stream>>>(weights, Rws, Iws);
  qc_main_kernel<<<Brows / ROWS_PER_BLOCK, BLK_THREADS,
                   LDS_FLOATS * sizeof(float), stream>>>(x, Wm, b, Rws, Iws, out);
}